// Decoder_60627758350697
// MI455X (gfx1250) — compile-verified
//
#include <hip/hip_runtime.h>
#include <hip/hip_bf16.h>

// ---------------------------------------------------------------------------
// Decoder step, B=32, S=2048, H=1024, V=32000, L=2.  fp32 throughout.
//  * Attention projection algebraically folded: q = h1 @ W_attn (b_attn
//    cancels in softmax) -> kills the 137-GFLOP / 256-MB proj GEMM.
//  * Attention over S fused flash-style: enc streamed from DRAM ONCE
//    (score sweep), context sweep re-reads the 256KB chunk L2-hot.
//  * All GEMMs on V_WMMA_F32_16X16X4_F32, one wave32 per 16x16 tile.
// DRAM budget ~= enc 256MB + W_out 262MB + weights 63MB -> ~25us @ 23.3TB/s.
// ---------------------------------------------------------------------------

typedef __attribute__((ext_vector_type(2))) float v2f;
typedef __attribute__((ext_vector_type(8))) float v8f;

#define Bq   32
#define Hq   1024
#define Sq   2048
#define Vq   32000
#define CHK  64            // s-rows per attention chunk
#define NCH  (Sq / CHK)    // 32 chunks per batch row

// ---- WMMA f32 16x16x4 helpers ---------------------------------------------
// A fragment (16x4): lane p=lane&15, hi=lane>>4:
//   a.x=A[m0+p][k+2*hi], a.y=A[m0+p][k+2*hi+1]
// B fragment (4x16): b.x=B[k+2*hi][n0+p], b.y=B[k+2*hi+1][n0+p]
// C/D (16x16 f32): vgpr i -> row m0 + hi*8 + i, col n0 + p.

// C += A(MxK) * Wt(NxK)^T      (B[k][n] = Wt[n][k]; Wt row-major, ld=ldw)
__device__ inline v8f wmma_acc_ABt(const float* __restrict__ A, int lda,
                                   const float* __restrict__ Wt, int ldw,
                                   int m0, int n0, int K, v8f acc, int lane) {
    const int p = lane & 15, hi = lane >> 4;
    const float* __restrict__ arow = A  + (size_t)(m0 + p) * lda + 2 * hi;
    const float* __restrict__ wrow = Wt + (size_t)(n0 + p) * ldw + 2 * hi;
#pragma unroll 4
    for (int k = 0; k < K; k += 4) {
        v2f a; a.x = arow[k]; a.y = arow[k + 1];
        v2f b; b.x = wrow[k]; b.y = wrow[k + 1];
        acc = __builtin_amdgcn_wmma_f32_16x16x4_f32(false, a, false, b,
                                                    (short)0, acc, false, false);
    }
    return acc;
}

// C += A(MxK) * W(KxN)         (B[k][n] = W[k][n]; W row-major, ld=ldw)
__device__ inline v8f wmma_acc_AB(const float* __restrict__ A, int lda,
                                  const float* __restrict__ W, int ldw,
                                  int m0, int n0, int K, v8f acc, int lane) {
    const int p = lane & 15, hi = lane >> 4;
    const float* __restrict__ arow = A + (size_t)(m0 + p) * lda + 2 * hi;
    const float* __restrict__ wcol = W + n0 + p;
#pragma unroll 4
    for (int k = 0; k < K; k += 4) {
        v2f a; a.x = arow[k]; a.y = arow[k + 1];
        v2f b; b.x = wcol[(size_t)(k + 2 * hi) * ldw];
               b.y = wcol[(size_t)(k + 2 * hi + 1) * ldw];
        acc = __builtin_amdgcn_wmma_f32_16x16x4_f32(false, a, false, b,
                                                    (short)0, acc, false, false);
    }
    return acc;
}

__device__ inline void wmma_store(float* __restrict__ C, int ldc,
                                  int m0, int n0, v8f acc, int lane) {
    const int p = lane & 15, hi = lane >> 4;
#pragma unroll
    for (int i = 0; i < 8; ++i)
        C[(size_t)(m0 + hi * 8 + i) * ldc + n0 + p] = acc[i];
}

__device__ inline float sigm(float x) { return 1.f / (1.f + expf(-x)); }

// ---- 1) embedding + concat with last_context -> rnn_in [B, 2H] -------------
__global__ void k_embed(const int* __restrict__ ids,
                        const float* __restrict__ emb,
                        const float* __restrict__ lastc,
                        float* __restrict__ rnn_in) {
    const int b = blockIdx.x;
    const int id = ids[b];
    for (int i = threadIdx.x; i < Hq; i += blockDim.x) {
        rnn_in[(size_t)b * 2 * Hq + i]      = emb[(size_t)id * Hq + i];
        rnn_in[(size_t)b * 2 * Hq + Hq + i] = lastc[(size_t)b * Hq + i];
    }
}

// ---- 2) GRU cell: 6 WMMA accumulations + fused gate nonlinearity -----------
// grid = (H/16, 2), block = 32 (one wave per 16x16 (b,j) tile, EXEC all ones)
__global__ void k_gru(const float* __restrict__ x, int Kx,
                      const float* __restrict__ hprev,
                      const float* __restrict__ Wih,   // [3H x Kx]
                      const float* __restrict__ Whh,   // [3H x H]
                      const float* __restrict__ bih,
                      const float* __restrict__ bhh,
                      float* __restrict__ hout,        // [B x H]
                      float* __restrict__ cat2,        // optional extra copy
                      int cat_ld, int cat_off) {
    const int lane = threadIdx.x;
    const int n0 = blockIdx.x * 16;       // gate column j
    const int m0 = blockIdx.y * 16;       // batch row b
    const v8f Z = {0.f, 0.f, 0.f, 0.f, 0.f, 0.f, 0.f, 0.f};

    v8f gir = wmma_acc_ABt(x, Kx, Wih,                       Kx, m0, n0, Kx, Z, lane);
    v8f giz = wmma_acc_ABt(x, Kx, Wih + (size_t)Hq * Kx,     Kx, m0, n0, Kx, Z, lane);
    v8f gin = wmma_acc_ABt(x, Kx, Wih + (size_t)2 * Hq * Kx, Kx, m0, n0, Kx, Z, lane);
    v8f ghr = wmma_acc_ABt(hprev, Hq, Whh,                       Hq, m0, n0, Hq, Z, lane);
    v8f ghz = wmma_acc_ABt(hprev, Hq, Whh + (size_t)Hq * Hq,     Hq, m0, n0, Hq, Z, lane);
    v8f ghn = wmma_acc_ABt(hprev, Hq, Whh + (size_t)2 * Hq * Hq, Hq, m0, n0, Hq, Z, lane);

    const int p = lane & 15, hi = lane >> 4;
    const int j = n0 + p;
#pragma unroll
    for (int i = 0; i < 8; ++i) {
        const int b = m0 + hi * 8 + i;
        float r = sigm((gir[i] + bih[j])          + (ghr[i] + bhh[j]));
        float z = sigm((giz[i] + bih[Hq + j])     + (ghz[i] + bhh[Hq + j]));
        float n = tanhf((gin[i] + bih[2 * Hq + j]) + r * (ghn[i] + bhh[2 * Hq + j]));
        float hv = (1.f - z) * n + z * hprev[(size_t)b * Hq + j];
        hout[(size_t)b * Hq + j] = hv;
        if (cat2) cat2[(size_t)b * cat_ld + cat_off + j] = hv;
    }
}

// ---- 3) q = h1 @ W_attn  (folded attention projection) ---------------------
__global__ void k_q(const float* __restrict__ h1,
                    const float* __restrict__ Wattn,   // [H x H] row-major
                    float* __restrict__ q) {
    const int lane = threadIdx.x;
    const int n0 = blockIdx.x * 16, m0 = blockIdx.y * 16;
    const v8f Z = {0.f, 0.f, 0.f, 0.f, 0.f, 0.f, 0.f, 0.f};
    v8f acc = wmma_acc_AB(h1, Hq, Wattn, Hq, m0, n0, Hq, Z, lane);
    wmma_store(q, Hq, m0, n0, acc, lane);
}

// ---- 4) fused attention chunk: scores + local softmax + partial context ----
// grid = (NCH, B), block = 256 (8 waves). enc chunk (256KB) read from DRAM in
// sweep 1, re-read L2-hot in sweep 2 -> enc crosses HBM only once overall.
__global__ void k_attn_part(const float* __restrict__ enc,
                            const float* __restrict__ q,
                            float* __restrict__ attn_raw,  // exp(score - m_c)
                            float* __restrict__ part,      // [B][NCH][H]
                            float* __restrict__ cmax,
                            float* __restrict__ csum) {
    __shared__ float sc[CHK];
    const int b = blockIdx.y, c = blockIdx.x;
    const int s0 = c * CHK;
    const int t = threadIdx.x, lane = t & 31, w = t >> 5;

    // sweep 1: 8 waves x 8 rows -> 64 scores
    const float4* __restrict__ q4 = (const float4*)(q + (size_t)b * Hq);
#pragma unroll
    for (int r = 0; r < CHK / 8; ++r) {
        const int s = s0 + w * 8 + r;
        const float4* __restrict__ e4 =
            (const float4*)(enc + ((size_t)b * Sq + s) * Hq);
        float acc = 0.f;
#pragma unroll
        for (int i = lane; i < Hq / 4; i += 32) {
            float4 ev = e4[i], qv = q4[i];
            acc += ev.x * qv.x + ev.y * qv.y + ev.z * qv.z + ev.w * qv.w;
        }
#pragma unroll
        for (int off = 16; off; off >>= 1) acc += __shfl_xor(acc, off);
        if (lane == 0) sc[w * 8 + r] = acc;
    }
    __syncthreads();

    // chunk-local max (redundant per thread; 64 LDS broadcast reads)
    float m = -3.4e38f;
#pragma unroll
    for (int i = 0; i < CHK; ++i) m = fmaxf(m, sc[i]);
    __syncthreads();                       // everyone done reading raw scores
    if (t < CHK) sc[t] = expf(sc[t] - m);  // overwrite with exp(score - m_c)
    __syncthreads();
    float ssum = 0.f;
#pragma unroll
    for (int i = 0; i < CHK; ++i) ssum += sc[i];

    if (t < CHK) attn_raw[(size_t)b * Sq + s0 + t] = sc[t];
    if (t == 0) { cmax[b * NCH + c] = m; csum[b * NCH + c] = ssum; }

    // sweep 2: partial context; thread owns 4 h-lanes (t, t+256, t+512, t+768)
    float a0 = 0.f, a1 = 0.f, a2 = 0.f, a3 = 0.f;
    const float* __restrict__ e = enc + ((size_t)b * Sq + s0) * Hq + t;
    for (int i = 0; i < CHK; ++i) {
        const float wgt = sc[i];
        const float* __restrict__ er = e + (size_t)i * Hq;
        a0 += wgt * er[0];
        a1 += wgt * er[256];
        a2 += wgt * er[512];
        a3 += wgt * er[768];
    }
    float* __restrict__ pp = part + ((size_t)b * NCH + c) * Hq + t;
    pp[0]   = a0;
    pp[256] = a1;
    pp[512] = a2;
    pp[768] = a3;
}

// ---- 5) per-b global softmax statistics over the NCH chunks ----------------
__global__ void k_attn_scale(const float* __restrict__ cmax,
                             const float* __restrict__ csum,
                             float* __restrict__ scale) {
    const int b = threadIdx.x;   // one lane per batch row, <<<1,32>>>
    float M = -3.4e38f;
#pragma unroll
    for (int c = 0; c < NCH; ++c) M = fmaxf(M, cmax[b * NCH + c]);
    float T = 0.f;
#pragma unroll
    for (int c = 0; c < NCH; ++c) T += csum[b * NCH + c] * expf(cmax[b * NCH + c] - M);
    const float invT = 1.f / T;
#pragma unroll
    for (int c = 0; c < NCH; ++c) scale[b * NCH + c] = expf(cmax[b * NCH + c] - M) * invT;
}

// ---- 6) normalize attention weights in place -------------------------------
__global__ void k_attn_norm(float* __restrict__ attn,
                            const float* __restrict__ scale) {
    const int b = blockIdx.x;
    for (int s = threadIdx.x; s < Sq; s += 256)
        attn[(size_t)b * Sq + s] *= scale[b * NCH + (s >> 6)];
}

// ---- 7) combine partial contexts -> context[b,h] ---------------------------
__global__ void k_ctx_combine(const float* __restrict__ part,
                              const float* __restrict__ scale,
                              float* __restrict__ ctx,
                              float* __restrict__ cat2) {
    const int b = blockIdx.y;
    const int h = blockIdx.x * 256 + threadIdx.x;
    float acc = 0.f;
#pragma unroll 4
    for (int c = 0; c < NCH; ++c)
        acc += scale[b * NCH + c] * part[((size_t)b * NCH + c) * Hq + h];
    ctx[(size_t)b * Hq + h] = acc;
    cat2[(size_t)b * 2 * Hq + Hq + h] = acc;
}

// ---- 8) logits = cat @ W_out^T + b_out  (WMMA, streams 262 MB of W_out) ----
__global__ void k_logits(const float* __restrict__ cat,
                         const float* __restrict__ Wout,  // [V x 2H]
                         const float* __restrict__ bout,
                         float* __restrict__ logits) {
    const int lane = threadIdx.x;
    const int n0 = blockIdx.x * 16, m0 = blockIdx.y * 16;
    const v8f Z = {0.f, 0.f, 0.f, 0.f, 0.f, 0.f, 0.f, 0.f};
    v8f acc = wmma_acc_ABt(cat, 2 * Hq, Wout, 2 * Hq, m0, n0, 2 * Hq, Z, lane);
    const int p = lane & 15, hi = lane >> 4;
    const float bv = bout[n0 + p];
#pragma unroll
    for (int i = 0; i < 8; ++i)
        logits[(size_t)(m0 + hi * 8 + i) * Vq + n0 + p] = acc[i] + bv;
}

// ---- 9) softmax over V, in place (block per b, 3 passes via L2) ------------
__global__ void k_softmaxV(float* __restrict__ logits) {
    __shared__ float red[256];
    const int b = blockIdx.x, t = threadIdx.x;
    float* row = logits + (size_t)b * Vq;
    float m = -3.4e38f;
    for (int i = t; i < Vq; i += 256) m = fmaxf(m, row[i]);
    red[t] = m; __syncthreads();
    for (int o = 128; o; o >>= 1) { if (t < o) red[t] = fmaxf(red[t], red[t + o]); __syncthreads(); }
    m = red[0]; __syncthreads();
    float sum = 0.f;
    for (int i = t; i < Vq; i += 256) sum += expf(row[i] - m);
    red[t] = sum; __syncthreads();
    for (int o = 128; o; o >>= 1) { if (t < o) red[t] += red[t + o]; __syncthreads(); }
    const float inv = 1.f / red[0];
    for (int i = t; i < Vq; i += 256) row[i] = expf(row[i] - m) * inv;
}

// ---------------------------------------------------------------------------
extern "C" void kernel_launch(void* const* d_in, const int* in_sizes, int n_in,
                              void* d_out, int out_size, void* d_ws, size_t ws_size,
                              hipStream_t stream) {
    (void)in_sizes; (void)n_in; (void)out_size; (void)ws_size;

    const int*   ids    = (const int*)d_in[0];
    const float* lastc  = (const float*)d_in[1];
    const float* hidden = (const float*)d_in[2];   // [2,B,H]
    const float* enc    = (const float*)d_in[3];   // [B,S,H]
    const float* emb    = (const float*)d_in[4];
    const float* Wattn  = (const float*)d_in[5];
    /* d_in[6] = b_attn: constant shift per b across s -> cancels in softmax */
    const float* Wih0 = (const float*)d_in[7];
    const float* Whh0 = (const float*)d_in[8];
    const float* bih0 = (const float*)d_in[9];
    const float* bhh0 = (const float*)d_in[10];
    const float* Wih1 = (const float*)d_in[11];
    const float* Whh1 = (const float*)d_in[12];
    const float* bih1 = (const float*)d_in[13];
    const float* bhh1 = (const float*)d_in[14];
    const float* Wout = (const float*)d_in[15];
    const float* bout = (const float*)d_in[16];

    // d_out layout (flat, return order): output | context | new_hidden | attn
    float* out    = (float*)d_out;
    float* output = out;                             // [32,32000]
    float* ctx    = out + (size_t)Bq * Vq;           // [32,1024]
    float* h0     = ctx + (size_t)Bq * Hq;           // new_hidden[0]
    float* h1     = h0  + (size_t)Bq * Hq;           // new_hidden[1]
    float* attn   = h1  + (size_t)Bq * Hq;           // [32,2048]

    float* wsf    = (float*)d_ws;
    float* rnn_in = wsf;                              // [32,2048]
    float* q      = rnn_in + (size_t)Bq * 2 * Hq;     // [32,1024]
    float* cat    = q      + (size_t)Bq * Hq;         // [32,2048]
    float* part   = cat    + (size_t)Bq * 2 * Hq;     // [32,NCH,1024] (4MB)
    float* cmax   = part   + (size_t)Bq * NCH * Hq;   // [32,NCH]
    float* csum   = cmax   + (size_t)Bq * NCH;        // [32,NCH]
    float* scale  = csum   + (size_t)Bq * NCH;        // [32,NCH]

    const float* hid0 = hidden;
    const float* hid1 = hidden + (size_t)Bq * Hq;

    k_embed<<<Bq, 256, 0, stream>>>(ids, emb, lastc, rnn_in);

    k_gru<<<dim3(Hq / 16, 2), 32, 0, stream>>>(rnn_in, 2 * Hq, hid0,
                                               Wih0, Whh0, bih0, bhh0,
                                               h0, nullptr, 0, 0);
    k_gru<<<dim3(Hq / 16, 2), 32, 0, stream>>>(h0, Hq, hid1,
                                               Wih1, Whh1, bih1, bhh1,
                                               h1, cat, 2 * Hq, 0);

    k_q<<<dim3(Hq / 16, 2), 32, 0, stream>>>(h1, Wattn, q);

    k_attn_part<<<dim3(NCH, Bq), 256, 0, stream>>>(enc, q, attn, part, cmax, csum);
    k_attn_scale<<<1, 32, 0, stream>>>(cmax, csum, scale);
    k_attn_norm<<<Bq, 256, 0, stream>>>(attn, scale);
    k_ctx_combine<<<dim3(Hq / 256, Bq), 256, 0, stream>>>(part, scale, ctx, cat);

    k_logits<<<dim3(Vq / 16, 2), 32, 0, stream>>>(cat, Wout, bout, output);
    k_softmaxV<<<Bq, 256, 0, stream>>>(output);
}